// MemoryLayerAttention_27917287424099
// MI455X (gfx1250) — compile-verified
//
#include <hip/hip_runtime.h>
#include <math.h>

typedef __attribute__((ext_vector_type(2))) float v2f;
typedef __attribute__((ext_vector_type(8))) float v8f;

// Problem constants (from reference): DIM=16, EMB=64, ROWS=64, BQ=128,
// Wx is (64, 4736); needed column bases: zi=0, zg=2368, zo=3552 (1024 each).

__device__ __forceinline__ float sigmoidf_(float x) { return 1.0f / (1.0f + expf(-x)); }

// Positional encoding PE[l][i] (L=65, EMB=64), computed on the fly.
__device__ __forceinline__ float pe_val(int l, int i) {
  float e = (float)(2 * (i >> 1)) * (1.0f / 64.0f);
  float denom = powf(10000.0f, e);
  float ang = (float)l / denom;
  return (i & 1) ? cosf(ang) : sinf(ang);
}

// 7 scan steps * DSTEPS(2) = 14 Euler steps of the 2-scalar potential ODE
// starting from (p0,p1)=(0,1). Only p0 (channel 0) feeds the memory.
__device__ float compute_p0() {
  const float C0 = 0.07915332f, C1 = 1.0334609f, C2 = 1.3365093f, C3 = 0.4505964f;
  const float M0 = 0.5f, M1 = 0.07879465f, M2 = 0.06618887f;
  const float T0 = 1.5931877f, T1 = 1.4378392f;
  const float PART = 1.5573331f / 2.0f;
  float p0 = 0.0f, p1 = 1.0f;
  for (int it = 0; it < 14; ++it) {
    float a = p0, b = p1;
    float d0 = C0 * sigmoidf_(100.0f * (0.0f - M0)) * (T0 - a)
             + C1 * sigmoidf_(100.0f * (a - M1)) * (T1 - a)
             + C2 * sigmoidf_(100.0f * (b - M2)) * (0.0f - a)
             + C3 * (0.0f - a);                      // sigmoid(inf)=1, TGT3=0
    float d1 = C0 * sigmoidf_(100.0f * (0.0f - M0)) * (T0 - b)
             + C1 * sigmoidf_(100.0f * (b - M1)) * (T1 - b)
             + C2 * sigmoidf_(100.0f * (a - M2)) * (0.0f - b)
             + C3 * (0.0f - b);
    p0 = a + d0 * PART;
    p1 = b + d1 * PART;
  }
  return p0;
}

// ---------------------------------------------------------------------------
// Kernel 1: batch-independent prep.
//   e_mem = p0 * colsum(Wm) + bm                         (64)
//   aug[l] = e_mem + PE[l],  l = 1..64                   (64 x 64, LDS)
//   Kmem[l-1][c] = bk[c] + aug[l] . Wk[:, c]             (64 x 128)
//   Vmem[l-1][c] = bv[c] + aug[l] . Wv[:, c]             (64 x 128)
// ---------------------------------------------------------------------------
__global__ void prep_kernel(const float* __restrict__ Wm, const float* __restrict__ bm,
                            const float* __restrict__ Wk, const float* __restrict__ bk,
                            const float* __restrict__ Wv, const float* __restrict__ bv,
                            float* __restrict__ Kmem, float* __restrict__ Vmem) {
  __shared__ float s_emem[64];
  __shared__ float s_aug[64 * 64];
  int tid = threadIdx.x;  // 256 threads
  float p0 = compute_p0();
  if (tid < 64) {
    float s = 0.0f;
    for (int r = 0; r < 256; ++r) s += Wm[r * 64 + tid];
    s_emem[tid] = p0 * s + bm[tid];
  }
  __syncthreads();
  for (int t = tid; t < 64 * 64; t += 256) {
    int l = 1 + (t >> 6), d = t & 63;
    s_aug[t] = s_emem[d] + pe_val(l, d);
  }
  __syncthreads();
  for (int t = tid; t < 64 * 128; t += 256) {
    int l = t >> 7, c = t & 127;
    float aK = bk[c], aV = bv[c];
    const float* arow = &s_aug[l * 64];
#pragma unroll 8
    for (int d = 0; d < 64; ++d) {
      float a = arow[d];
      aK = fmaf(a, Wk[d * 128 + c], aK);
      aV = fmaf(a, Wv[d * 128 + c], aV);
    }
    Kmem[t] = aK;
    Vmem[t] = aV;
  }
}

// ---------------------------------------------------------------------------
// Kernel 2: per-batch 65-way attention -> O[b] (64). One block per b (128).
// ---------------------------------------------------------------------------
__global__ void attn_kernel(const float* __restrict__ queries, const float* __restrict__ values,
                            const float* __restrict__ Wi, const float* __restrict__ bi,
                            const float* __restrict__ Wq, const float* __restrict__ bq,
                            const float* __restrict__ Wk, const float* __restrict__ bk,
                            const float* __restrict__ Wv, const float* __restrict__ bv,
                            const float* __restrict__ Wo, const float* __restrict__ bo,
                            const float* __restrict__ Kmem, const float* __restrict__ Vmem,
                            float* __restrict__ O) {
  __shared__ float x[32], a0[64], q[128], k0[128], v0[128], att[130], ctx[128];
  int b = blockIdx.x, tid = threadIdx.x;  // 128 threads
  // x_7[b] = [queries[b/16, b%16, :], values[b/16, 7, :]]
  if (tid < 16) x[tid] = queries[b * 16 + tid];
  else if (tid < 32) x[tid] = values[(b >> 4) * 128 + 112 + (tid - 16)];
  __syncthreads();
  if (tid < 64) {
    float s = bi[tid];
#pragma unroll 8
    for (int j = 0; j < 32; ++j) s = fmaf(x[j], Wi[j * 64 + tid], s);
    a0[tid] = s + pe_val(0, tid);
  }
  __syncthreads();
  {
    int c = tid;  // c = h*64 + k
    float sq = bq[c], sk = bk[c], sv = bv[c];
#pragma unroll 8
    for (int d = 0; d < 64; ++d) {
      float a = a0[d];
      sq = fmaf(a, Wq[d * 128 + c], sq);
      sk = fmaf(a, Wk[d * 128 + c], sk);
      sv = fmaf(a, Wv[d * 128 + c], sv);
    }
    q[c] = sq; k0[c] = sk; v0[c] = sv;
  }
  __syncthreads();
  for (int t = tid; t < 130; t += 128) {
    int h = t / 65, l = t % 65;
    const float* qh = &q[h * 64];
    float s = 0.0f;
    if (l == 0) {
#pragma unroll 8
      for (int d = 0; d < 64; ++d) s = fmaf(qh[d], k0[h * 64 + d], s);
    } else {
      const float* kr = &Kmem[(l - 1) * 128 + h * 64];
#pragma unroll 8
      for (int d = 0; d < 64; ++d) s = fmaf(qh[d], kr[d], s);
    }
    att[t] = s * 0.125f;  // 1/sqrt(EMB)
  }
  __syncthreads();
  if (tid < 2) {
    float m = -1e30f;
    for (int l = 0; l < 65; ++l) m = fmaxf(m, att[tid * 65 + l]);
    float sum = 0.0f;
    for (int l = 0; l < 65; ++l) { float e = expf(att[tid * 65 + l] - m); att[tid * 65 + l] = e; sum += e; }
    float inv = 1.0f / sum;
    for (int l = 0; l < 65; ++l) att[tid * 65 + l] *= inv;
  }
  __syncthreads();
  {
    int c = tid, h = c >> 6;
    float s = att[h * 65] * v0[c];
#pragma unroll 8
    for (int l = 1; l < 65; ++l) s = fmaf(att[h * 65 + l], Vmem[(l - 1) * 128 + c], s);
    ctx[c] = s;
  }
  __syncthreads();
  if (tid < 64) {
    float s = bo[tid];
#pragma unroll 8
    for (int c = 0; c < 128; ++c) s = fmaf(ctx[c], Wo[c * 64 + tid], s);
    O[b * 64 + tid] = s;
  }
}

// ---------------------------------------------------------------------------
// Kernel 3: Z = O(128x64) @ Wx columns {zi,zg,zo}, fused LSTM-gate activation.
// One wave (32 threads) per 16x16 output tile; 3 accumulators share one A.
// Uses V_WMMA_F32_16X16X4_F32 (pure-f32 matrix pipe, 16 K-steps per tile).
// A layout: lanes 0-15 K={0,1}, lanes 16-31 K={2,3}; B mirrors the K split;
// C: lane holds col (lane&15), VGPR j holds row j + 8*(lane>=16).
// ---------------------------------------------------------------------------
__global__ void __launch_bounds__(32)
gemm_act_kernel(const float* __restrict__ O, const float* __restrict__ Wx,
                const float* __restrict__ bl, float* __restrict__ out) {
  int m = blockIdx.x >> 6;   // 0..7   (row tile of 128)
  int n = blockIdx.x & 63;   // 0..63  (col tile of 1024)
  int lane = threadIdx.x;
  int rsel = lane & 15;
  int hi = lane >> 4;        // 0 or 1
  int rA = m * 16 + rsel;
  int coli = 0    + n * 16 + rsel;   // zi slice of Wx/bl
  int colg = 2368 + n * 16 + rsel;   // zg slice
  int colo = 3552 + n * 16 + rsel;   // zo slice

  v8f ci = {}, cg = {}, co = {};
  const float* Arow = &O[rA * 64 + 2 * hi];
  for (int k4 = 0; k4 < 64; k4 += 4) {
    v2f a;
    a.x = Arow[k4];
    a.y = Arow[k4 + 1];
    const float* W0 = &Wx[(k4 + 2 * hi) * 4736];
    const float* W1 = W0 + 4736;
    v2f bi_; bi_.x = W0[coli]; bi_.y = W1[coli];
    v2f bg_; bg_.x = W0[colg]; bg_.y = W1[colg];
    v2f bo_; bo_.x = W0[colo]; bo_.y = W1[colo];
    ci = __builtin_amdgcn_wmma_f32_16x16x4_f32(false, a, false, bi_, (short)0, ci, false, false);
    cg = __builtin_amdgcn_wmma_f32_16x16x4_f32(false, a, false, bg_, (short)0, cg, false, false);
    co = __builtin_amdgcn_wmma_f32_16x16x4_f32(false, a, false, bo_, (short)0, co, false, false);
  }
  float bli = bl[coli], blg = bl[colg], blo = bl[colo];
  int rowbase = m * 16 + hi * 8;
  int col = n * 16 + rsel;
#pragma unroll
  for (int j = 0; j < 8; ++j) {
    float zi = ci[j] + bli, zg = cg[j] + blg, zo = co[j] + blo;
    float c = sigmoidf_(zi) * tanhf(zg);
    float h = sigmoidf_(zo) * tanhf(c);
    out[(rowbase + j) * 1024 + col] = h;
  }
}

extern "C" void kernel_launch(void* const* d_in, const int* in_sizes, int n_in,
                              void* d_out, int out_size, void* d_ws, size_t ws_size,
                              hipStream_t stream) {
  const float* queries = (const float*)d_in[0];
  const float* values  = (const float*)d_in[1];
  const float* Wi = (const float*)d_in[2];
  const float* bi = (const float*)d_in[3];
  const float* Wm = (const float*)d_in[4];
  const float* bm = (const float*)d_in[5];
  const float* Wq = (const float*)d_in[6];
  const float* bq = (const float*)d_in[7];
  const float* Wk = (const float*)d_in[8];
  const float* bk = (const float*)d_in[9];
  const float* Wv = (const float*)d_in[10];
  const float* bv = (const float*)d_in[11];
  const float* Wo = (const float*)d_in[12];
  const float* bo = (const float*)d_in[13];
  const float* Wx = (const float*)d_in[14];
  const float* bl = (const float*)d_in[15];

  float* ws   = (float*)d_ws;
  float* Kmem = ws;            // 64*128
  float* Vmem = ws + 8192;     // 64*128
  float* O    = ws + 16384;    // 128*64
  float* out  = (float*)d_out; // 128*1024

  prep_kernel<<<dim3(1), dim3(256), 0, stream>>>(Wm, bm, Wk, bk, Wv, bv, Kmem, Vmem);
  attn_kernel<<<dim3(128), dim3(128), 0, stream>>>(queries, values, Wi, bi, Wq, bq,
                                                   Wk, bk, Wv, bv, Wo, bo, Kmem, Vmem, O);
  gemm_act_kernel<<<dim3(512), dim3(32), 0, stream>>>(O, Wx, bl, out);
}